// GNN_1_57097295233371
// MI455X (gfx1250) — compile-verified
//
#include <hip/hip_runtime.h>

// ---------------------------------------------------------------------------
// GNN block for MI455X (gfx1250, wave32).
// FP32 WMMA (v_wmma_f32_16x16x4_f32) — problem is latency/L2 bound (~3.4
// GFLOP, ~5 MB resident), so keep full fp32 precision; fuse each 3-layer MLP
// into one kernel with hidden activations resident in LDS. Branch-free inner
// loops (EXEC stays all-ones), dual accumulator chains per wave.
// ---------------------------------------------------------------------------

typedef __attribute__((ext_vector_type(2))) float v2f;
typedef __attribute__((ext_vector_type(8))) float v8f;

#define GNN_B   128
#define GNN_K   32
#define GNN_C   64
#define GNN_H   512
#define LDS_STRIDE 516   // 516 % 64 == 4 -> conflict-free strided row reads

__device__ __forceinline__ v8f wmma_f32(v2f a, v2f b, v8f c) {
    // D = A(16x4) * B(4x16) + C(16x16), all fp32
    return __builtin_amdgcn_wmma_f32_16x16x4_f32(
        /*neg_a=*/false, a, /*neg_b=*/false, b,
        /*c_mod=*/(short)0, c, /*reuse_a=*/false, /*reuse_b=*/false);
}

// Build XA[4096,64] = concat(x, a) rows
__global__ void gnn_xa_kernel(const float* __restrict__ x,
                              const float* __restrict__ a,
                              float* __restrict__ XA) {
    int idx = blockIdx.x * blockDim.x + threadIdx.x;     // 4096*64
    if (idx >= GNN_B * GNN_K * GNN_C) return;
    int row = idx >> 6;          // b*32 + k
    int c   = idx & 63;
    float v = (c < 48) ? x[row * 48 + c] : a[row * 16 + (c - 48)];
    XA[idx] = v;
}

// Build E[1024,128]: row e=(i*32+j) = concat(XA[i], XA[j])  (batch-0 nodes!)
__global__ void gnn_edge_input_kernel(const float* __restrict__ XA,
                                      float* __restrict__ E) {
    int idx = blockIdx.x * blockDim.x + threadIdx.x;     // 1024*128
    if (idx >= GNN_K * GNN_K * 2 * GNN_C) return;
    int e = idx >> 7;
    int c = idx & 127;
    int i = e >> 5, j = e & 31;
    int src = (c < GNN_C) ? (i * GNN_C + c) : (j * GNN_C + (c - GNN_C));
    E[idx] = XA[src];
}

// esum[32,64] = sum_j Ee[i*32+j][c]
__global__ void gnn_reduce_kernel(const float* __restrict__ Ee,
                                  float* __restrict__ esum) {
    int idx = blockIdx.x * blockDim.x + threadIdx.x;     // 32*64
    if (idx >= GNN_K * GNN_C) return;
    int i = idx >> 6, c = idx & 63;
    float s = 0.f;
    #pragma unroll 8
    for (int j = 0; j < GNN_K; ++j)
        s += Ee[(i * GNN_K + j) * GNN_C + c];
    esum[idx] = s;
}

// XR[4096,64] = XA + esum[row%32]  (residual + broadcast edge sum)
__global__ void gnn_residual_kernel(const float* __restrict__ XA,
                                    const float* __restrict__ esum,
                                    float* __restrict__ XR) {
    int idx = blockIdx.x * blockDim.x + threadIdx.x;     // 4096*64
    if (idx >= GNN_B * GNN_K * GNN_C) return;
    int row = idx >> 6;
    int c   = idx & 63;
    XR[idx] = XA[idx] + esum[(row & 31) * GNN_C + c];
}

// ---------------------------------------------------------------------------
// Fused 3-layer MLP:  out[M,64] = (relu(relu(Xin@w1+b1)@w2+b2))@w3+b3
//   Xin: [M, K0], w1:[K0,512], w2:[512,512], w3:[512,64]
// One workgroup (4 wave32s) owns 16 rows; hidden [16,512] tiles in LDS.
// Layers 1/2: each wave processes two N-tiles at once -> one shared
// A-fragment load feeds two independent WMMA accumulator chains.
// No divergent branches anywhere -> EXEC all-ones for every WMMA.
// ---------------------------------------------------------------------------
__global__ __launch_bounds__(128) void gnn_fused_mlp3_kernel(
    const float* __restrict__ Xin, int K0,
    const float* __restrict__ w1, const float* __restrict__ b1,
    const float* __restrict__ w2, const float* __restrict__ b2,
    const float* __restrict__ w3, const float* __restrict__ b3,
    float* __restrict__ out) {

    __shared__ float h1[16 * LDS_STRIDE];
    __shared__ float h2[16 * LDS_STRIDE];

    const int lane = threadIdx.x & 31;
    const int wave = threadIdx.x >> 5;      // 0..3
    const int half = lane >> 4;             // 0..1
    const int lr   = lane & 15;             // 0..15
    const int m0   = blockIdx.x * 16;

    // Warm L2/WGP$ for the weight matrices every workgroup re-reads
    // (lowers to global_prefetch_b8).
    __builtin_prefetch(w2 + (size_t)threadIdx.x * 16, 0, 1);
    __builtin_prefetch(w3 + (size_t)threadIdx.x * 16, 0, 1);

    // ----- layer 1: Xin[16 x K0] @ w1[K0 x 512] -> relu -> h1 -----
    const float* arow = Xin + (size_t)(m0 + lr) * K0;

    for (int t = wave; t < GNN_H / 16; t += 8) {
        const int n0a = t * 16;
        const int n0b = (t + 4) * 16;
        const float bva = b1[n0a + lr];
        const float bvb = b1[n0b + lr];
        v8f acc0, acc1;
        #pragma unroll
        for (int r = 0; r < 8; ++r) { acc0[r] = bva; acc1[r] = bvb; }

        for (int k0 = 0; k0 < K0; k0 += 4) {
            const int ka = k0 + 2 * half;
            v2f a;
            a.x = arow[ka];
            a.y = arow[ka + 1];
            const float* bpa = w1 + (size_t)ka * GNN_H + n0a + lr;
            const float* bpb = w1 + (size_t)ka * GNN_H + n0b + lr;
            v2f b0, b1v;
            b0.x  = bpa[0];  b0.y  = bpa[GNN_H];
            b1v.x = bpb[0];  b1v.y = bpb[GNN_H];
            acc0 = wmma_f32(a, b0,  acc0);
            acc1 = wmma_f32(a, b1v, acc1);
        }
        #pragma unroll
        for (int r = 0; r < 8; ++r) {
            float v0 = acc0[r], v1 = acc1[r];
            h1[(r + 8 * half) * LDS_STRIDE + n0a + lr] = v0 > 0.f ? v0 : 0.f;
            h1[(r + 8 * half) * LDS_STRIDE + n0b + lr] = v1 > 0.f ? v1 : 0.f;
        }
    }
    __syncthreads();

    // ----- layer 2: h1[16 x 512] @ w2[512 x 512] -> relu -> h2 -----
    for (int t = wave; t < GNN_H / 16; t += 8) {
        const int n0a = t * 16;
        const int n0b = (t + 4) * 16;
        const float bva = b2[n0a + lr];
        const float bvb = b2[n0b + lr];
        v8f acc0, acc1;
        #pragma unroll
        for (int r = 0; r < 8; ++r) { acc0[r] = bva; acc1[r] = bvb; }

        for (int k0 = 0; k0 < GNN_H; k0 += 4) {
            const int ka = k0 + 2 * half;
            v2f a;
            a.x = h1[lr * LDS_STRIDE + ka];
            a.y = h1[lr * LDS_STRIDE + ka + 1];
            const float* bpa = w2 + (size_t)ka * GNN_H + n0a + lr;
            const float* bpb = w2 + (size_t)ka * GNN_H + n0b + lr;
            v2f b0, b1v;
            b0.x  = bpa[0];  b0.y  = bpa[GNN_H];
            b1v.x = bpb[0];  b1v.y = bpb[GNN_H];
            acc0 = wmma_f32(a, b0,  acc0);
            acc1 = wmma_f32(a, b1v, acc1);
        }
        #pragma unroll
        for (int r = 0; r < 8; ++r) {
            float v0 = acc0[r], v1 = acc1[r];
            h2[(r + 8 * half) * LDS_STRIDE + n0a + lr] = v0 > 0.f ? v0 : 0.f;
            h2[(r + 8 * half) * LDS_STRIDE + n0b + lr] = v1 > 0.f ? v1 : 0.f;
        }
    }
    __syncthreads();

    // ----- layer 3: h2[16 x 512] @ w3[512 x 64] -> out (no relu) -----
    {
        const int n0 = wave * 16;           // 4 waves cover 64 cols
        const float bv = b3[n0 + lr];
        v8f acc;
        #pragma unroll
        for (int r = 0; r < 8; ++r) acc[r] = bv;

        for (int k0 = 0; k0 < GNN_H; k0 += 4) {
            const int ka = k0 + 2 * half;
            v2f a;
            a.x = h2[lr * LDS_STRIDE + ka];
            a.y = h2[lr * LDS_STRIDE + ka + 1];
            const float* bp = w3 + (size_t)ka * GNN_C + n0 + lr;
            v2f b;
            b.x = bp[0];
            b.y = bp[GNN_C];
            acc = wmma_f32(a, b, acc);
        }
        #pragma unroll
        for (int r = 0; r < 8; ++r)
            out[(size_t)(m0 + r + 8 * half) * GNN_C + n0 + lr] = acc[r];
    }
}

extern "C" void kernel_launch(void* const* d_in, const int* in_sizes, int n_in,
                              void* d_out, int out_size, void* d_ws, size_t ws_size,
                              hipStream_t stream) {
    const float* x   = (const float*)d_in[0];
    const float* a   = (const float*)d_in[1];
    const float* ew1 = (const float*)d_in[2];
    const float* eb1 = (const float*)d_in[3];
    const float* ew2 = (const float*)d_in[4];
    const float* eb2 = (const float*)d_in[5];
    const float* ew3 = (const float*)d_in[6];
    const float* eb3 = (const float*)d_in[7];
    const float* nw1 = (const float*)d_in[8];
    const float* nb1 = (const float*)d_in[9];
    const float* nw2 = (const float*)d_in[10];
    const float* nb2 = (const float*)d_in[11];
    const float* nw3 = (const float*)d_in[12];
    const float* nb3 = (const float*)d_in[13];

    float* ws   = (float*)d_ws;
    float* XA   = ws;                                // [4096, 64]
    float* E    = XA + GNN_B * GNN_K * GNN_C;        // [1024, 128]
    float* Ee   = E  + GNN_K * GNN_K * 2 * GNN_C;    // [1024, 64]
    float* esum = Ee + GNN_K * GNN_K * GNN_C;        // [32, 64]
    float* XR   = esum + GNN_K * GNN_C;              // [4096, 64]
    // total scratch: ~2.8 MB

    // 1) xa = concat(x, a)
    gnn_xa_kernel<<<(GNN_B * GNN_K * GNN_C + 255) / 256, 256, 0, stream>>>(x, a, XA);
    // 2) edge-pair input matrix (batch-0 nodes only, per reference semantics)
    gnn_edge_input_kernel<<<(GNN_K * GNN_K * 2 * GNN_C + 255) / 256, 256, 0, stream>>>(XA, E);
    // 3) edge MLP (fused 3 layers), M = 1024
    gnn_fused_mlp3_kernel<<<GNN_K * GNN_K / 16, 128, 0, stream>>>(
        E, 2 * GNN_C, ew1, eb1, ew2, eb2, ew3, eb3, Ee);
    // 4) esum[i] = sum_j Ee[i,j]
    gnn_reduce_kernel<<<(GNN_K * GNN_C + 255) / 256, 256, 0, stream>>>(Ee, esum);
    // 5) residual: XR = XA + broadcast(esum)
    gnn_residual_kernel<<<(GNN_B * GNN_K * GNN_C + 255) / 256, 256, 0, stream>>>(XA, esum, XR);
    // 6) node MLP (fused 3 layers), M = 4096
    gnn_fused_mlp3_kernel<<<GNN_B * GNN_K / 16, 128, 0, stream>>>(
        XR, GNN_C, nw1, nb1, nw2, nb2, nw3, nb3, (float*)d_out);
}